// DynamicDepthwiseConv2d_73280732004404
// MI455X (gfx1250) — compile-verified
//
#include <hip/hip_runtime.h>

// ---------------------------------------------------------------------------
// DynamicDepthwiseConv2d fused kernel for gfx1250 (MI455X, wave32, WMMA)
//   stage 1: style MLP -> per-sample depthwise 3x3 kernels + biases (in d_ws)
//   stage 2: fused depthwise 3x3 (VALU) + 1x1 pointwise GEMM (v_wmma_f32_16x16x4_f32)
// ---------------------------------------------------------------------------

typedef __attribute__((ext_vector_type(2))) float v2f;
typedef __attribute__((ext_vector_type(8))) float v8f;

#define CIN   256
#define COUT  256
#define HIMG  128
#define WIMG  128
#define NB    8
#define KTAP  9
#define TPIX  64            // pixels per block tile (one row, 64 cols)
#define YSTR  (CIN + 4)     // padded LDS row stride (floats) -> conflict-free b64 B loads

// ------------------------- stage 1: style MLP ------------------------------
__global__ __launch_bounds__(256) void ddw_style_mlp(
    const float* __restrict__ style,   // [N, C]
    const float* __restrict__ Wk,      // [C*9, C]
    const float* __restrict__ bk,      // [C*9]
    const float* __restrict__ Wb,      // [C, C]
    const float* __restrict__ bb,      // [C]
    float* __restrict__ dkern,         // [N, C*9]
    float* __restrict__ dbias)         // [N, C]
{
  __shared__ float s[CIN];
  const int n = blockIdx.x;
  const int t = threadIdx.x;
  s[t] = style[n * CIN + t];
  __syncthreads();

  // dynamic depthwise kernels: 2304 outputs, 256 threads -> 9 each
  for (int j = t; j < CIN * KTAP; j += 256) {
    const float* wr = Wk + (size_t)j * CIN;
    float acc = bk[j];
    #pragma unroll 8
    for (int c = 0; c < CIN; ++c) acc += s[c] * wr[c];
    dkern[n * CIN * KTAP + j] = acc;
  }
  // dynamic biases: 256 outputs, 1 each
  {
    const float* wr = Wb + (size_t)t * CIN;
    float acc = bb[t];
    #pragma unroll 8
    for (int c = 0; c < CIN; ++c) acc += s[c] * wr[c];
    dbias[n * CIN + t] = acc;
  }
}

// ---------------- stage 2: fused depthwise + pointwise WMMA ----------------
__global__ __launch_bounds__(256) void ddw_fused_conv(
    const float* __restrict__ x,       // [N, C, H, W]
    const float* __restrict__ dkern,   // [N, C, 9]
    const float* __restrict__ dbias,   // [N, C]
    const float* __restrict__ Wp,      // [OC, C]
    const float* __restrict__ bp,      // [OC]
    float* __restrict__ out)           // [N, OC, H, W]
{
  __shared__ float ys[TPIX * YSTR];    // y[pixel][channel], padded rows (~65 KB)

  const int bid = blockIdx.x;          // N * H * (W/TPIX) = 8*128*2 = 2048 blocks
  const int wt  = bid & 1;             // column tile
  const int h   = (bid >> 1) & (HIMG - 1);
  const int n   = bid >> 8;
  const int w0  = wt * TPIX;

  const int tid = threadIdx.x;

  // ---------------- depthwise 3x3 with dynamic per-(n,c) kernels ----------
  {
    const int p  = tid & (TPIX - 1);   // pixel within tile
    const int cg = tid >> 6;           // channel group 0..3 (64 channels each)
    const int col = w0 + p;
    const bool cl = (col - 1) >= 0;
    const bool cr = (col + 1) < WIMG;

    const float* xn = x + (size_t)n * CIN * HIMG * WIMG;
    const float* kn = dkern + (size_t)n * CIN * KTAP;
    const float* bn = dbias + (size_t)n * CIN;

    #pragma unroll 2
    for (int i = 0; i < 64; ++i) {
      const int c = cg * 64 + i;
      const float* kr = kn + c * KTAP;       // 9 taps (uniform per 64 lanes -> bcast)
      float acc = bn[c];
      const float* xc = xn + (size_t)c * HIMG * WIMG;
      #pragma unroll
      for (int dy = 0; dy < 3; ++dy) {
        const int r = h + dy - 1;
        if (r >= 0 && r < HIMG) {            // uniform branch (h is uniform)
          const float* xr = xc + (size_t)r * WIMG;
          const float xl = cl ? xr[col - 1] : 0.0f;
          const float xm = xr[col];
          const float xv = cr ? xr[col + 1] : 0.0f;
          acc += kr[dy * 3 + 0] * xl + kr[dy * 3 + 1] * xm + kr[dy * 3 + 2] * xv;
        }
      }
      ys[p * YSTR + c] = acc;
    }
  }
  __syncthreads();

  // ---------------- pointwise 1x1: out = Wp @ y via f32 WMMA --------------
  // Per wave: 2 OC tiles of 16; per OC tile: 4 accumulators (16-pixel tiles),
  // K-loop over C in steps of 4 -> v_wmma_f32_16x16x4_f32.
  {
    const int lane  = tid & 31;
    const int wv    = tid >> 5;          // wave 0..7
    const int lm    = lane & 15;
    const int khalf = (lane >> 4) << 1;  // lanes 16..31 carry K+2,K+3 halves

    #pragma unroll
    for (int oi = 0; oi < 2; ++oi) {
      const int oc0 = (wv * 2 + oi) * 16;
      const float* wrow = Wp + (size_t)(oc0 + lm) * CIN + khalf; // A row, 8B aligned
      __builtin_prefetch(wrow, 0, 1);    // pull Wp row toward the WGP

      v8f acc0 = {}, acc1 = {}, acc2 = {}, acc3 = {};

      for (int k = 0; k < CIN; k += 4) {
        // A fragment (16x4 f32): lane -> M=oc0+lm; VGPR0/1 -> K = k+khalf, k+khalf+1
        const v2f a = *(const v2f*)(wrow + k);
        // B fragments (4x16 f32): lane -> N=pixel; VGPR0/1 -> K = k+khalf, k+khalf+1
        const v2f b0 = *(const v2f*)(&ys[(0 * 16 + lm) * YSTR + k + khalf]);
        const v2f b1 = *(const v2f*)(&ys[(1 * 16 + lm) * YSTR + k + khalf]);
        const v2f b2 = *(const v2f*)(&ys[(2 * 16 + lm) * YSTR + k + khalf]);
        const v2f b3 = *(const v2f*)(&ys[(3 * 16 + lm) * YSTR + k + khalf]);
        acc0 = __builtin_amdgcn_wmma_f32_16x16x4_f32(false, a, false, b0, (short)0, acc0, false, false);
        acc1 = __builtin_amdgcn_wmma_f32_16x16x4_f32(false, a, false, b1, (short)0, acc1, false, false);
        acc2 = __builtin_amdgcn_wmma_f32_16x16x4_f32(false, a, false, b2, (short)0, acc2, false, false);
        acc3 = __builtin_amdgcn_wmma_f32_16x16x4_f32(false, a, false, b3, (short)0, acc3, false, false);
      }

      // D layout: VGPR v, lane -> row M = v + 8*(lane>=16), col N = lane&15
      const int moff = (lane >> 4) << 3;
      #pragma unroll
      for (int v = 0; v < 8; ++v) {
        const int oc = oc0 + moff + v;
        const float bpv = bp[oc];
        float* orow = out + (((size_t)(n * COUT + oc) * HIMG + h) * WIMG + w0);
        orow[0 * 16 + lm] = acc0[v] + bpv;
        orow[1 * 16 + lm] = acc1[v] + bpv;
        orow[2 * 16 + lm] = acc2[v] + bpv;
        orow[3 * 16 + lm] = acc3[v] + bpv;
      }
    }
  }
}

// ---------------------------------------------------------------------------
extern "C" void kernel_launch(void* const* d_in, const int* in_sizes, int n_in,
                              void* d_out, int out_size, void* d_ws, size_t ws_size,
                              hipStream_t stream) {
  (void)in_sizes; (void)n_in; (void)out_size; (void)ws_size;
  const float* x     = (const float*)d_in[0];
  const float* style = (const float*)d_in[1];
  const float* Wk    = (const float*)d_in[2];
  const float* bk    = (const float*)d_in[3];
  const float* Wb    = (const float*)d_in[4];
  const float* bb    = (const float*)d_in[5];
  const float* Wp    = (const float*)d_in[6];
  const float* bp    = (const float*)d_in[7];
  float* out = (float*)d_out;

  float* dkern = (float*)d_ws;                       // [N, C*9]
  float* dbias = dkern + (size_t)NB * CIN * KTAP;    // [N, C]

  ddw_style_mlp<<<NB, 256, 0, stream>>>(style, Wk, bk, Wb, bb, dkern, dbias);

  const int nblocks = NB * HIMG * (WIMG / TPIX);     // 2048
  ddw_fused_conv<<<nblocks, 256, 0, stream>>>(x, dkern, dbias, Wp, bp, out);
}